// MultiTypeEQParameterHead_48619029790865
// MI455X (gfx1250) — compile-verified
//
#include <hip/hip_runtime.h>
#include <math.h>

typedef __attribute__((ext_vector_type(2))) float v2f;
typedef __attribute__((ext_vector_type(8))) float v8f;

#define B_  4096
#define E_  512
#define M_  128
#define NB_ 5
#define H_  64
#define NT_ 5
#define L1P 134   // padded l1 pitch: 3 zero cols | 128 data | 3 zero cols

__constant__ float LOGFMIN_C[5] = {2.9957323f, 2.9957323f, 6.9077553f, 2.9957323f, 7.6009025f};
__constant__ float LOGFMAX_C[5] = {9.9034876f, 8.5171932f, 9.9034876f, 6.2146081f, 9.9034876f};
#define LOG_F_MIN_ 2.9957323f
#define LOG_F_MAX_ 9.9034876f

__device__ __forceinline__ float gelu_f(float x) {
    return 0.5f * x * (1.0f + erff(x * 0.70710678118654752f));
}
__device__ __forceinline__ float clampf(float x, float lo, float hi) {
    return fminf(fmaxf(x, lo), hi);
}
__device__ __forceinline__ v8f wmma_f32(v2f a, v2f b, v8f c) {
    // D = A(16x4) * B(4x16) + C(16x16), full fp32 on the matrix pipe
    return __builtin_amdgcn_wmma_f32_16x16x4_f32(false, a, false, b, (short)0, c, false, false);
}

// ---------------------------------------------------------------------------
// Kernel 1: trunk = relu(embedding @ trunk_w + trunk_b)   (4096x512)@(512x320)
// grid = (5, 256); block = 128 (4 waves); wave w does col-tile blockIdx.x*4+w
// ---------------------------------------------------------------------------
__global__ __launch_bounds__(128) void trunk_kernel(
    const float* __restrict__ emb, const float* __restrict__ W,
    const float* __restrict__ bias, float* __restrict__ trunk)
{
    __shared__ float As[16 * 516];  // padded pitch to spread LDS banks
    const int tid  = threadIdx.x;
    const int wave = tid >> 5;
    const int lane = tid & 31;
    const int half = lane >> 4;
    const int mrow = lane & 15;
    const int row0 = blockIdx.y * 16;
    const int col0 = (blockIdx.x * 4 + wave) * 16;

    for (int i = tid; i < 16 * 512; i += 128) {
        int r = i >> 9, k = i & 511;
        As[r * 516 + k] = emb[(row0 + r) * E_ + k];
    }
    __syncthreads();

    v8f acc = {};
    for (int k0 = 0; k0 < E_; k0 += 4) {
        v2f a, b;
        a.x = As[mrow * 516 + k0 + 2 * half];
        a.y = As[mrow * 516 + k0 + 2 * half + 1];
        b.x = W[(k0 + 2 * half) * 320 + col0 + mrow];
        b.y = W[(k0 + 2 * half + 1) * 320 + col0 + mrow];
        acc = wmma_f32(a, b, acc);
    }
#pragma unroll
    for (int r = 0; r < 8; ++r) {
        int row = row0 + r + 8 * half;
        int col = col0 + mrow;
        trunk[row * 320 + col] = fmaxf(acc[r] + bias[col], 0.0f);
    }
}

// ---------------------------------------------------------------------------
// Kernel 2: full _band_ctx for one branch. grid = B, block = 128 (4 waves).
// lc2 conv as fully-unrolled WMMA GEMM over zero-padded LDS (no branches in
// the matrix loop); vals einsum via WMMA.
// ---------------------------------------------------------------------------
__global__ __launch_bounds__(128) void band_ctx_kernel(
    const float* __restrict__ mel, const float* __restrict__ trunk,
    const float* __restrict__ gc1_w, const float* __restrict__ gc1_b,
    const float* __restrict__ gc2_w, const float* __restrict__ gc2_b,
    const float* __restrict__ glin_w, const float* __restrict__ glin_b,
    const float* __restrict__ lc1_w, const float* __restrict__ lc1_b,
    const float* __restrict__ lc2_w, const float* __restrict__ lc2_b,
    const float* __restrict__ qw,   const float* __restrict__ qb,
    const float* __restrict__ vw,   const float* __restrict__ vb,
    const float* __restrict__ ln_g, const float* __restrict__ ln_b,
    float* __restrict__ ctx_out)
{
    __shared__ float mel_c[M_];          // 128
    __shared__ float trunk_s[NB_ * H_];  // 320
    __shared__ float l1buf[32 * L1P];    // 16.75 KB : padded lc1 output; g2 reuse
    __shared__ float l2t[64 * 16];       // 4 KB  : lc2 tile / g-conv1 reuse
    __shared__ float vals[M_ * H_];      // 32 KB : vals[m][o]
    __shared__ float gvec[H_];
    __shared__ float qs[NB_ * H_];
    __shared__ float sc[NB_ * M_];
    __shared__ float feat[NB_ * H_];
    __shared__ float red[128];

    const int b    = blockIdx.x;
    const int tid  = threadIdx.x;
    const int wave = tid >> 5;
    const int lane = tid & 31;
    const int half = lane >> 4;
    const int nlan = lane & 15;

    // ---- phase 0: mel centering, trunk row ----
    float mv = mel[b * M_ + tid];
    red[tid] = mv;
    __syncthreads();
    for (int s = 64; s > 0; s >>= 1) { if (tid < s) red[tid] += red[tid + s]; __syncthreads(); }
    float mmean = red[0] * (1.0f / 128.0f);
    __syncthreads();
    mel_c[tid] = mv - mmean;
    for (int i = tid; i < NB_ * H_; i += 128) trunk_s[i] = trunk[b * 320 + i];
    __syncthreads();

    // ---- phase 1: lc1 conv (32ch x 128, k=15, pad 7) + gelu, into padded l1 ----
    for (int i = tid; i < 32 * L1P; i += 128) {
        int ic = i / L1P, c = i - ic * L1P;
        float v = 0.0f;
        if (c >= 3 && c < 131) {
            int m = c - 3;
            float s = lc1_b[ic];
#pragma unroll
            for (int k = 0; k < 15; ++k) {
                int p = m + k - 7;
                if (p >= 0 && p < 128) s += lc1_w[ic * 15 + k] * mel_c[p];
            }
            v = gelu_f(s);
        }
        l1buf[i] = v;
    }
    __syncthreads();

    // ---- phase 2: lc2 conv-as-GEMM (K=224 im2col) + vals GEMM, per 16-col tile
    for (int mt = 0; mt < 8; ++mt) {
        const int m0 = mt * 16;
        {   // lc2 tile: D(oc x m) 16x16 per wave, K = 32*7, fully unrolled
            const int oc0  = wave * 16;
            const int bcol = m0 + nlan;  // padded l1 col for dk=0 (= m0+nlan+dk-3+3)
            v8f acc = {};
#pragma unroll
            for (int kk = 0; kk < 224; kk += 4) {
                const int k0 = kk + 2 * half;
                const int ic0 = k0 / 7, dk0 = k0 - ic0 * 7;
                const int k1 = k0 + 1;
                const int ic1 = k1 / 7, dk1 = k1 - ic1 * 7;
                v2f a, bb;
                a.x  = lc2_w[(oc0 + nlan) * 224 + k0];
                a.y  = lc2_w[(oc0 + nlan) * 224 + k1];
                bb.x = l1buf[ic0 * L1P + bcol + dk0];
                bb.y = l1buf[ic1 * L1P + bcol + dk1];
                acc = wmma_f32(a, bb, acc);
            }
#pragma unroll
            for (int r = 0; r < 8; ++r) {
                int oc = oc0 + r + 8 * half;
                l2t[oc * 16 + nlan] = gelu_f(acc[r] + lc2_b[oc]);
            }
        }
        __syncthreads();
        {   // vals tile: D(o x m) 16x16 per wave, K = 64, fully unrolled
            const int o0 = wave * 16;
            v8f acc = {};
#pragma unroll
            for (int kk = 0; kk < 64; kk += 4) {
                v2f a, bb;
                a.x  = vw[(o0 + nlan) * 64 + kk + 2 * half];
                a.y  = vw[(o0 + nlan) * 64 + kk + 2 * half + 1];
                bb.x = l2t[(kk + 2 * half) * 16 + nlan];
                bb.y = l2t[(kk + 2 * half + 1) * 16 + nlan];
                acc = wmma_f32(a, bb, acc);
            }
#pragma unroll
            for (int r = 0; r < 8; ++r) {
                int o = o0 + r + 8 * half;
                int m = m0 + nlan;
                vals[m * H_ + o] = acc[r] + vb[o];
            }
        }
        __syncthreads();
    }

    // ---- phase 3: g-branch (small convs, VALU). g1 in l2t; g2 reuses l1buf. ----
    float* g2buf = l1buf;  // l1 dead after phase 2; reuse first 4 KB
    for (int i = tid; i < 32 * 32; i += 128) {
        int oc = i >> 5, p = i & 31;
        float s = gc1_b[oc];
#pragma unroll
        for (int k = 0; k < 15; ++k) {
            int x = 4 * p + k - 7;
            if (x >= 0 && x < 128) s += gc1_w[oc * 15 + k] * mel_c[x];
        }
        l2t[oc * 32 + p] = gelu_f(s);
    }
    __syncthreads();
    for (int i = tid; i < 64 * 16; i += 128) {
        int oc = i >> 4, p = i & 15;
        float s = gc2_b[oc];
        for (int ic = 0; ic < 32; ++ic) {
#pragma unroll
            for (int k = 0; k < 7; ++k) {
                int x = 2 * p + k - 3;
                if (x >= 0 && x < 32) s += gc2_w[(oc * 32 + ic) * 7 + k] * l2t[ic * 32 + x];
            }
        }
        g2buf[oc * 16 + p] = gelu_f(s);
    }
    __syncthreads();
    if (tid < 64) {
        float s = 0.0f;
        for (int p = 0; p < 16; ++p) s += g2buf[tid * 16 + p];
        red[tid] = s * (1.0f / 16.0f);
    }
    __syncthreads();
    if (tid < 64) {
        float s = glin_b[tid];
        for (int ic = 0; ic < 64; ++ic) s += red[ic] * glin_w[ic * 64 + tid];
        gvec[tid] = gelu_f(s);
    }
    __syncthreads();

    // ---- phase 4: q, attention, softmax, feat, layernorm, +g ----
    for (int i = tid; i < NB_ * H_; i += 128) {
        int n = i >> 6, c = i & 63;
        float s = qb[c];
        for (int k = 0; k < 64; ++k) s += trunk_s[n * 64 + k] * qw[k * 64 + c];
        qs[i] = s;
    }
    __syncthreads();
    {
        int m = tid;  // 128 threads = 128 positions
        for (int n = 0; n < NB_; ++n) {
            float s = 0.0f;
            for (int c = 0; c < 64; ++c) s += qs[n * 64 + c] * vals[m * H_ + c];
            sc[n * M_ + m] = s * 0.125f;  // 1/sqrt(64)
        }
    }
    __syncthreads();
    for (int n = 0; n < NB_; ++n) {
        float v = sc[n * M_ + tid];
        red[tid] = v;
        __syncthreads();
        for (int s = 64; s > 0; s >>= 1) { if (tid < s) red[tid] = fmaxf(red[tid], red[tid + s]); __syncthreads(); }
        float mx = red[0];
        __syncthreads();
        float e = expf(v - mx);
        red[tid] = e;
        __syncthreads();
        for (int s = 64; s > 0; s >>= 1) { if (tid < s) red[tid] += red[tid + s]; __syncthreads(); }
        float sum = red[0];
        __syncthreads();
        sc[n * M_ + tid] = e / sum;
    }
    if (tid < 64) {
        for (int n = 0; n < NB_; ++n) {
            float s = 0.0f;
            for (int m = 0; m < 128; ++m) s += sc[n * M_ + m] * vals[m * H_ + tid];
            feat[n * H_ + tid] = s;
        }
    }
    __syncthreads();
    for (int n = 0; n < NB_; ++n) {
        float v = (tid < 64) ? feat[n * H_ + tid] : 0.0f;
        red[tid] = v;
        __syncthreads();
        for (int s = 64; s > 0; s >>= 1) { if (tid < s) red[tid] += red[tid + s]; __syncthreads(); }
        float mu = red[0] * (1.0f / 64.0f);
        __syncthreads();
        float d = (tid < 64) ? (v - mu) : 0.0f;
        red[tid] = d * d;
        __syncthreads();
        for (int s = 64; s > 0; s >>= 1) { if (tid < s) red[tid] += red[tid + s]; __syncthreads(); }
        float var = red[0] * (1.0f / 64.0f);
        __syncthreads();
        if (tid < 64) {
            float o = (v - mu) * rsqrtf(var + 1e-5f) * ln_g[tid] + ln_b[tid] + gvec[tid];
            ctx_out[(b * NB_ + n) * H_ + tid] = o;
        }
    }
}

// ---------------------------------------------------------------------------
// Kernel 3: heads. grid = B, block = 128.
// ---------------------------------------------------------------------------
__global__ __launch_bounds__(128) void heads_kernel(
    const float* __restrict__ mel, const float* __restrict__ trunk,
    const float* __restrict__ tctx_g, const float* __restrict__ gctx_g,
    const float* __restrict__ t_fuse_w, const float* __restrict__ t_fuse_b,
    const float* __restrict__ g_fuse_w, const float* __restrict__ g_fuse_b,
    const float* __restrict__ fctx_w, const float* __restrict__ fctx_b,
    const float* __restrict__ freq_w, const float* __restrict__ freq_b,
    const float* __restrict__ gain_head_w, const float* __restrict__ gain_head_b,
    const float* __restrict__ q_w, const float* __restrict__ q_b,
    const float* __restrict__ th1_w, const float* __restrict__ th1_b,
    const float* __restrict__ th2_w, const float* __restrict__ th2_b,
    const float* __restrict__ th3_w, const float* __restrict__ th3_b,
    float* __restrict__ out)
{
    __shared__ float tr[320], tc[320], gc[320], melr[128], shelf[3];
    __shared__ float ti[320], gi[320], fi[320];
    __shared__ float h1[640], h2[320], lg[32], pb[32];

    const int b = blockIdx.x, tid = threadIdx.x;
    for (int i = tid; i < 320; i += 128) {
        tr[i] = trunk[b * 320 + i];
        tc[i] = tctx_g[b * 320 + i];
        gc[i] = gctx_g[b * 320 + i];
    }
    melr[tid] = mel[b * M_ + tid];
    __syncthreads();

    if (tid == 0) {  // shelf features (serial, tiny)
        float sa16 = 0.f, sl16 = 0.f, stot = 0.f, msum = 0.f;
        for (int i = 0; i < 128; ++i) {
            float a = fabsf(melr[i]);
            stot += a;
            if (i < 16)   sa16 += a;
            if (i >= 112) sl16 += a;
            msum += melr[i];
        }
        float lo = logf((sa16 / 16.f + 1e-6f) / ((stot - sa16) / 112.f + 1e-6f));
        float hi = logf((sl16 / 16.f + 1e-6f) / ((stot - sl16) / 112.f + 1e-6f));
        float mm = msum / 128.f;
        float num = 0.f, den = 0.f;
        for (int i = 0; i < 128; ++i) {
            float x = -1.f + 2.f * (float)i / 127.f;
            num += x * (melr[i] - mm);
            den += x * x;
        }
        shelf[0] = clampf(lo, -5.f, 5.f);
        shelf[1] = clampf(hi, -5.f, 5.f);
        shelf[2] = clampf(num / (den + 1e-8f), -5.f, 5.f);
    }
    // fused: type_in / gain_ctx / fctx matvecs (each (5x128)@(128x64))
    for (int i = tid; i < 320; i += 128) {
        int n = i >> 6, c = i & 63;
        float st = t_fuse_b[c], sg = g_fuse_b[c], sf = fctx_b[c];
        for (int k = 0; k < 64; ++k) {
            float tk = tr[n * 64 + k];
            st += tk * t_fuse_w[k * 64 + c];
            sg += tk * g_fuse_w[k * 64 + c];
            sf += tk * fctx_w[k * 64 + c];
            float tck = tc[n * 64 + k], gck = gc[n * 64 + k];
            st += tck * t_fuse_w[(64 + k) * 64 + c];
            sg += gck * g_fuse_w[(64 + k) * 64 + c];
            sf += tck * fctx_w[(64 + k) * 64 + c];
        }
        ti[i] = gelu_f(st);
        gi[i] = gelu_f(sg);
        fi[i] = gelu_f(sf);
    }
    __syncthreads();

    for (int i = tid; i < 640; i += 128) {  // th1: (5x67)@(67x128)
        int n = i >> 7, j = i & 127;
        float s = th1_b[j];
        for (int k = 0; k < 64; ++k) s += ti[n * 64 + k] * th1_w[k * 128 + j];
        s += shelf[0] * th1_w[64 * 128 + j];
        s += shelf[1] * th1_w[65 * 128 + j];
        s += shelf[2] * th1_w[66 * 128 + j];
        h1[i] = fmaxf(s, 0.f);
    }
    __syncthreads();
    for (int i = tid; i < 320; i += 128) {  // th2
        int n = i >> 6, j = i & 63;
        float s = th2_b[j];
        for (int k = 0; k < 128; ++k) s += h1[n * 128 + k] * th2_w[k * 64 + j];
        h2[i] = fmaxf(s, 0.f);
    }
    __syncthreads();
    if (tid < 25) {  // th3 logits
        int n = tid / 5, t = tid % 5;
        float s = th3_b[t];
        for (int k = 0; k < 64; ++k) s += h2[n * 64 + k] * th3_w[k * 5 + t];
        lg[tid] = s;
    }
    __syncthreads();
    if (tid < NB_) {  // softmax over NT per band
        float mx = -1e30f;
        for (int t = 0; t < NT_; ++t) mx = fmaxf(mx, lg[tid * 5 + t]);
        float sum = 0.f;
        for (int t = 0; t < NT_; ++t) { float e = expf(lg[tid * 5 + t] - mx); pb[tid * 5 + t] = e; sum += e; }
        for (int t = 0; t < NT_; ++t) pb[tid * 5 + t] /= sum;
    }
    __syncthreads();
    if (tid < NB_) {
        int n = tid;
        float gdot = gain_head_b[0];
        for (int c = 0; c < 64; ++c) gdot += gi[n * 64 + c] * gain_head_w[c];
        float gainful = pb[n * 5 + 0] + pb[n * 5 + 1] + pb[n * 5 + 2];
        float gain = clampf(gdot, -12.f, 12.f) * gainful;

        float fdot = freq_b[0];
        for (int c = 0; c < 64; ++c) fdot += fi[n * 64 + c] * freq_w[c];
        float fnorm = 1.f / (1.f + expf(-fdot));
        float lfmin = 0.f, lfmax = 0.f;
        for (int t = 0; t < NT_; ++t) { lfmin += pb[n * 5 + t] * LOGFMIN_C[t]; lfmax += pb[n * 5 + t] * LOGFMAX_C[t]; }
        float log_f = clampf(lfmin + fnorm * (lfmax - lfmin), LOG_F_MIN_, LOG_F_MAX_);
        float freq = expf(log_f);

        float qdot = q_b[0];
        for (int c = 0; c < 64; ++c) qdot += tr[n * 64 + c] * q_w[c];
        float sp = (qdot > 20.f) ? qdot : log1pf(expf(qdot));
        float qv = clampf(sp + 0.1f, 0.1f, 10.f);

        float* o = out + (size_t)(b * NB_ + n) * 8;
        o[0] = gain; o[1] = freq; o[2] = qv;
        for (int t = 0; t < NT_; ++t) o[3 + t] = lg[n * 5 + t];
    }
}

// ---------------------------------------------------------------------------
extern "C" void kernel_launch(void* const* d_in, const int* in_sizes, int n_in,
                              void* d_out, int out_size, void* d_ws, size_t ws_size,
                              hipStream_t stream) {
    (void)in_sizes; (void)n_in; (void)out_size; (void)ws_size;
    int idx = 0;
    const float* emb        = (const float*)d_in[idx++];
    const float* mel        = (const float*)d_in[idx++];
    const float* trunk_w    = (const float*)d_in[idx++];
    const float* trunk_b    = (const float*)d_in[idx++];
    const float* gain_head_w= (const float*)d_in[idx++];
    const float* gain_head_b= (const float*)d_in[idx++];
    const float* fctx_w     = (const float*)d_in[idx++];
    const float* fctx_b     = (const float*)d_in[idx++];
    const float* freq_w     = (const float*)d_in[idx++];
    const float* freq_b     = (const float*)d_in[idx++];
    const float* q_w        = (const float*)d_in[idx++];
    const float* q_b        = (const float*)d_in[idx++];
    const float* tw[18]; for (int j = 0; j < 18; ++j) tw[j] = (const float*)d_in[idx++];
    const float* gw[18]; for (int j = 0; j < 18; ++j) gw[j] = (const float*)d_in[idx++];
    const float* th1_w = (const float*)d_in[idx++];
    const float* th1_b = (const float*)d_in[idx++];
    const float* th2_w = (const float*)d_in[idx++];
    const float* th2_b = (const float*)d_in[idx++];
    const float* th3_w = (const float*)d_in[idx++];
    const float* th3_b = (const float*)d_in[idx++];

    float* ws    = (float*)d_ws;
    float* trunk = ws;                       // 4096*320
    float* tctx  = ws + (size_t)B_ * 320;    // 4096*5*64
    float* gctx  = tctx + (size_t)B_ * 320;  // 4096*5*64
    float* outp  = (float*)d_out;

    trunk_kernel<<<dim3(5, 256), 128, 0, stream>>>(emb, trunk_w, trunk_b, trunk);

    band_ctx_kernel<<<B_, 128, 0, stream>>>(mel, trunk,
        tw[0], tw[1], tw[2], tw[3], tw[4], tw[5], tw[6], tw[7], tw[8], tw[9],
        tw[10], tw[11], tw[12], tw[13], tw[14], tw[15], tctx);

    band_ctx_kernel<<<B_, 128, 0, stream>>>(mel, trunk,
        gw[0], gw[1], gw[2], gw[3], gw[4], gw[5], gw[6], gw[7], gw[8], gw[9],
        gw[10], gw[11], gw[12], gw[13], gw[14], gw[15], gctx);

    heads_kernel<<<B_, 128, 0, stream>>>(mel, trunk, tctx, gctx,
        tw[16], tw[17], gw[16], gw[17], fctx_w, fctx_b, freq_w, freq_b,
        gain_head_w, gain_head_b, q_w, q_b,
        th1_w, th1_b, th2_w, th2_b, th3_w, th3_b, outp);
}